// GPT2Editor_53695681135066
// MI455X (gfx1250) — compile-verified
//
#include <hip/hip_runtime.h>

typedef __attribute__((ext_vector_type(2))) float v2f;
typedef __attribute__((ext_vector_type(8))) float v8f;

#define EDIM 2048
#define FDIM 4
#define SDIM 4096
#define KSL  8

// ---------------------------------------------------------------------------
// K1: v partials.  v[f,e] = sum_k h_last[k] * Wc[f,k,E+e]
// grid = F(4) * eblocks(8) * kslices(8) = 256 blocks of 256 threads.
// ---------------------------------------------------------------------------
__global__ void kv_partial(const float* __restrict__ hidden,
                           const float* __restrict__ Wc,
                           float* __restrict__ v_part) {
  __shared__ float hs[256];
  const int b  = blockIdx.x;
  const int ks = b & 7;          // k slice (256 k's each)
  const int eb = (b >> 3) & 7;   // e block (256 e's each)
  const int f  = b >> 6;         // 0..3
  const int t  = threadIdx.x;
  hs[t] = hidden[127 * EDIM + ks * 256 + t];   // h_last slice -> LDS
  __syncthreads();
  const int e = eb * 256 + t;
  const float* wp = Wc + (size_t)f * EDIM * 2 * EDIM
                       + (size_t)(ks * 256) * (2 * EDIM) + EDIM + e;
  float acc = 0.f;
  #pragma unroll 4
  for (int k = 0; k < 256; ++k)
    acc = fmaf(hs[k], wp[(size_t)k * (2 * EDIM)], acc);
  v_part[(size_t)ks * (FDIM * EDIM) + f * EDIM + e] = acc;
}

// K1b: v[f,e] = sum_slices + bc[f, E+e]
__global__ void kv_reduce(const float* __restrict__ v_part,
                          const float* __restrict__ bc,
                          float* __restrict__ v_ws) {
  const int idx = blockIdx.x * 256 + threadIdx.x;   // 0..8191
  const int f = idx >> 11;
  const int e = idx & 2047;
  float acc = bc[f * 2 * EDIM + EDIM + e];
  #pragma unroll
  for (int s = 0; s < KSL; ++s) acc += v_part[s * (FDIM * EDIM) + idx];
  v_ws[idx] = acc;
}

// ---------------------------------------------------------------------------
// K2: r partials on the matrix pipe.
// r = vcat(1x8192) @ Wpcat(8192x2048), padded to 16 rows for
// V_WMMA_F32_16X16X4_F32.  grid = 128 column tiles (16 cols each),
// 8 waves/block, wave wib owns K-slice [wib*1024, wib*1024+1024).
// A 16x4 layout: (M = lane%16, K = 2*laneHi + comp); only M==0 nonzero —
// enforced by a 0/1 mask multiply (no EXEC-predicated LDS loads).
// B 4x16 layout: (K = 2*laneHi + comp, N = lane%16), software-pipelined.
// ---------------------------------------------------------------------------
__global__ void r_partial_wmma(const float* __restrict__ v_ws,
                               const float* __restrict__ Wp,
                               float* __restrict__ r_part) {
  __shared__ float vl[FDIM * EDIM];            // 32 KB of 320 KB/WGP
  const int t = threadIdx.x;
  for (int i = t; i < FDIM * EDIM; i += 256) vl[i] = v_ws[i];
  __syncthreads();

  const int lane = t & 31;
  const int wib  = t >> 5;                      // 0..7 -> K slice
  const int ln   = lane & 15;                   // N / M index
  const int hi   = lane >> 4;                   // lane half
  const int o0   = blockIdx.x * 16;
  const int kbase = wib * 1024;
  const int kend  = kbase + 1024;
  const float m  = (ln == 0) ? 1.f : 0.f;       // row-0 mask (A padding)

  const float* wpc = Wp + o0 + ln;              // Wp flat: [kk*2048 + o]
  const float* vk  = vl + 2 * hi;               // lane-half base into LDS

  v8f c = {};
  // Prologue: B fragment for the first step.
  v2f b0;
  b0.x = wpc[(size_t)(kbase + 2 * hi    ) * EDIM];
  b0.y = wpc[(size_t)(kbase + 2 * hi + 1) * EDIM];

  #pragma unroll 4
  for (int k4 = kbase; k4 < kend - 4; k4 += 4) {
    // Prefetch next step's B fragment before issuing this step's WMMA.
    v2f bn;
    bn.x = wpc[(size_t)(k4 + 4 + 2 * hi    ) * EDIM];
    bn.y = wpc[(size_t)(k4 + 4 + 2 * hi + 1) * EDIM];
    // A fragment: unconditional contiguous LDS pair, masked to row 0.
    v2f a;
    a.x = m * vk[k4];
    a.y = m * vk[k4 + 1];
    c = __builtin_amdgcn_wmma_f32_16x16x4_f32(
            false, a, false, b0, (short)0, c, false, false);
    b0 = bn;
  }
  // Epilogue: last step (no further prefetch, no OOB read past Wp).
  {
    v2f a;
    a.x = m * vk[kend - 4];
    a.y = m * vk[kend - 3];
    c = __builtin_amdgcn_wmma_f32_16x16x4_f32(
            false, a, false, b0, (short)0, c, false, false);
  }
  // D row M=0 = C-VGPR0 on lanes 0..15 (N = lane).
  if (hi == 0) r_part[wib * EDIM + o0 + ln] = c[0];
}

// K2b: r[o] = sum_slices + sum_f bp[f,o]
__global__ void r_reduce(const float* __restrict__ r_part,
                         const float* __restrict__ bp,
                         float* __restrict__ r_ws) {
  const int o = blockIdx.x * 256 + threadIdx.x;   // 0..2047
  float acc = 0.f;
  #pragma unroll
  for (int f = 0; f < FDIM; ++f) acc += bp[f * EDIM + o];
  #pragma unroll
  for (int s = 0; s < KSL; ++s) acc += r_part[s * EDIM + o];
  r_ws[o] = acc;
}

// K3: out[s, :] = r for all 4096 rows (float4 stores, r hot in L2).
__global__ void bcast_rows(const float* __restrict__ r_ws,
                           float4* __restrict__ out) {
  const int s = blockIdx.x;
  const int t = threadIdx.x;
  const float4* r4 = (const float4*)r_ws;       // 512 float4 per row
  const float4 x0 = r4[t];
  const float4 x1 = r4[t + 256];
  float4* row = out + (size_t)s * (EDIM / 4);
  row[t]       = x0;
  row[t + 256] = x1;
}

extern "C" void kernel_launch(void* const* d_in, const int* in_sizes, int n_in,
                              void* d_out, int out_size, void* d_ws, size_t ws_size,
                              hipStream_t stream) {
  (void)in_sizes; (void)n_in; (void)out_size; (void)ws_size;
  const float* hidden = (const float*)d_in[0];  // (1,128,2048)
  const float* Wc     = (const float*)d_in[4];  // (4,2048,4096)
  const float* bc     = (const float*)d_in[5];  // (4,4096)
  const float* Wp     = (const float*)d_in[6];  // (4,2048,2048)
  const float* bp     = (const float*)d_in[7];  // (4,2048)
  // (enc_hidden, Wq, bq are provably dead: softmax over a singleton axis == 1)

  float* ws     = (float*)d_ws;
  float* v_part = ws;                           // 8*8192 floats
  float* v_ws   = v_part + KSL * FDIM * EDIM;   // 8192
  float* r_part = v_ws   + FDIM * EDIM;         // 8*2048
  float* r_ws   = r_part + KSL * EDIM;          // 2048   (total ~369 KB)

  kv_partial    <<<256,  256, 0, stream>>>(hidden, Wc, v_part);
  kv_reduce     <<<32,   256, 0, stream>>>(v_part, bc, v_ws);
  r_partial_wmma<<<128,  256, 0, stream>>>(v_ws, Wp, r_part);
  r_reduce      <<<8,    256, 0, stream>>>(r_part, bp, r_ws);
  bcast_rows    <<<SDIM, 256, 0, stream>>>(r_ws, (float4*)d_out);
}